// Net_14053132993014
// MI455X (gfx1250) — compile-verified
//
#include <hip/hip_runtime.h>
#include <hip/hip_bf16.h>

#define N_NODES 100000
#define N_EDGES 1600000
#define C 128

typedef __attribute__((ext_vector_type(2))) float v2f;
typedef __attribute__((ext_vector_type(8))) float v8f;

// ---------------- degree / normalization ----------------

__global__ __launch_bounds__(256) void k_init_deg(unsigned* cnt) {
    int i = blockIdx.x * 256 + threadIdx.x;
    if (i < N_NODES) cnt[i] = 1u;  // self loop contributes 1
}

__global__ __launch_bounds__(256) void k_count_deg(const int* __restrict__ col,
                                                   unsigned* __restrict__ cnt) {
    int e = blockIdx.x * 256 + threadIdx.x;
    if (e < N_EDGES) atomicAdd(&cnt[col[e]], 1u);
}

__global__ __launch_bounds__(256) void k_rsqrt_deg(float* dis) {
    int i = blockIdx.x * 256 + threadIdx.x;
    if (i < N_NODES) {
        unsigned c = ((const unsigned*)dis)[i];
        dis[i] = rsqrtf((float)c);   // deg >= 1 always (self loop)
    }
}

// ---------------- fp32 WMMA GEMM: out[M,128] = (relu?)A[M,128] @ W.T (+bias) ----------------
// One wave computes a 16-row x 128-col strip. K looped in steps of 4 using
// V_WMMA_F32_16X16X4_F32 (full fp32 precision, matches fp32 reference).
//
// A fragment (16x4 f32): lanes 0-15 hold M=lane, K={k,k+1}; lanes 16-31 hold M=lane-16, K={k+2,k+3}
// B fragment (4x16 f32): lane n = N=lane&15, K pair selected by lane>>4; B[k][n] = W[n][k]
// C/D (16x16 f32): VGPR j: lanes 0-15 -> (M=j, N=lane); lanes 16-31 -> (M=j+8, N=lane-16)

__global__ __launch_bounds__(256) void k_gemm_wmma(const float* __restrict__ A,
                                                   const float* __restrict__ W,
                                                   float* __restrict__ out,
                                                   const float* __restrict__ bias,
                                                   int relu_in, int n_strips) {
    int wave = (int)((blockIdx.x * 256 + threadIdx.x) >> 5);
    int lane = threadIdx.x & 31;
    if (wave >= n_strips) return;          // wave-uniform: EXEC stays all-1s for WMMA

    const int m0    = wave * 16;
    const int lhalf = lane >> 4;           // 0 or 1
    const int l15   = lane & 15;

    const float* abase = A + (size_t)(m0 + l15) * C + lhalf * 2;
    const float* wbase = W + (size_t)l15 * C + lhalf * 2;

    v8f acc[8];
    v8f zero = {0.f, 0.f, 0.f, 0.f, 0.f, 0.f, 0.f, 0.f};
#pragma unroll
    for (int n = 0; n < 8; ++n) acc[n] = zero;

    for (int k = 0; k < C; k += 4) {
        v2f av = *(const v2f*)(abase + k);
        if (relu_in) {
            av.x = fmaxf(av.x, 0.0f);
            av.y = fmaxf(av.y, 0.0f);
        }
#pragma unroll
        for (int n = 0; n < 8; ++n) {
            v2f bv = *(const v2f*)(wbase + (size_t)n * 16 * C + k);
            acc[n] = __builtin_amdgcn_wmma_f32_16x16x4_f32(
                false, av, false, bv, (short)0, acc[n], false, false);
        }
    }

    const int rbase = m0 + lhalf * 8;
#pragma unroll
    for (int n = 0; n < 8; ++n) {
        int cidx = n * 16 + l15;
        float bv = bias ? bias[cidx] : 0.0f;
#pragma unroll
        for (int j = 0; j < 8; ++j) {
            out[(size_t)(rbase + j) * C + cidx] = acc[n][j] + bv;
        }
    }
}

// ---------------- aggregation ----------------
// init: out[i,c] = bias[c] + dis[i]^2 * h[i,c]   (self-loop term folded in)
__global__ __launch_bounds__(256) void k_agg_init(const float* __restrict__ h,
                                                  const float* __restrict__ dis,
                                                  const float* __restrict__ bias,
                                                  float* __restrict__ out) {
    int idx = blockIdx.x * 256 + threadIdx.x;   // N*C exactly divisible by 256
    int i = idx >> 7, c = idx & 127;
    float d = dis[i];
    out[idx] = bias[c] + d * d * h[idx];
}

// edges: one wave per edge, float4 per lane (32*4 = 128 channels)
__global__ __launch_bounds__(256) void k_agg_edges(const int* __restrict__ ei,
                                                   const float* __restrict__ h,
                                                   const float* __restrict__ dis,
                                                   float* __restrict__ out) {
    int t = blockIdx.x * 256 + threadIdx.x;
    int e = t >> 5;
    int lane = t & 31;
    if (e >= N_EDGES) return;
    int row = ei[e];             // source
    int col = ei[N_EDGES + e];   // target
    float norm = dis[row] * dis[col];
    const float4 hv = *(const float4*)(h + (size_t)row * C + lane * 4);
    float* op = out + (size_t)col * C + lane * 4;
    unsafeAtomicAdd(op + 0, norm * hv.x);   // native global_atomic_add_f32 in L2
    unsafeAtomicAdd(op + 1, norm * hv.y);
    unsafeAtomicAdd(op + 2, norm * hv.z);
    unsafeAtomicAdd(op + 3, norm * hv.w);
}

// ---------------- log softmax (wave32 per node) ----------------
__global__ __launch_bounds__(256) void k_logsoftmax(const float* __restrict__ logits,
                                                    float* __restrict__ out) {
    int node = (int)((blockIdx.x * 256 + threadIdx.x) >> 5);
    int lane = threadIdx.x & 31;
    if (node >= N_NODES) return;
    const float4 v = *(const float4*)(logits + (size_t)node * C + lane * 4);
    float m = fmaxf(fmaxf(v.x, v.y), fmaxf(v.z, v.w));
#pragma unroll
    for (int off = 16; off > 0; off >>= 1) m = fmaxf(m, __shfl_xor(m, off, 32));
    float s = expf(v.x - m) + expf(v.y - m) + expf(v.z - m) + expf(v.w - m);
#pragma unroll
    for (int off = 16; off > 0; off >>= 1) s += __shfl_xor(s, off, 32);
    float lse = m + logf(s);
    float4 o = {v.x - lse, v.y - lse, v.z - lse, v.w - lse};
    *(float4*)(out + (size_t)node * C + lane * 4) = o;
}

// ---------------- driver ----------------

extern "C" void kernel_launch(void* const* d_in, const int* in_sizes, int n_in,
                              void* d_out, int out_size, void* d_ws, size_t ws_size,
                              hipStream_t stream) {
    const float* x  = (const float*)d_in[0];
    const int*   ei = (const int*)d_in[1];     // [2, E] row-major: rows at [0,E), cols at [E,2E)
    const float* W1 = (const float*)d_in[2];
    const float* b1 = (const float*)d_in[3];
    const float* W2 = (const float*)d_in[4];
    const float* b2 = (const float*)d_in[5];
    const float* Wl = (const float*)d_in[6];
    const float* bl = (const float*)d_in[7];
    float* out = (float*)d_out;

    // workspace: dis[N] (padded to 100096 floats) + t[N*C]  ~= 51.6 MB
    float* dis = (float*)d_ws;
    float* t   = dis + 100096;

    const int strips      = N_NODES / 16;       // 6250 (exact)
    const int gemm_blocks = (strips + 7) / 8;   // 8 waves / block

    // symmetric normalization (shared by both convs)
    k_init_deg <<<(N_NODES + 255) / 256, 256, 0, stream>>>((unsigned*)dis);
    k_count_deg<<<(N_EDGES + 255) / 256, 256, 0, stream>>>(ei + N_EDGES, (unsigned*)dis);
    k_rsqrt_deg<<<(N_NODES + 255) / 256, 256, 0, stream>>>(dis);

    // layer 1: t = x @ W1.T ; out = b1 + self + scatter
    k_gemm_wmma<<<gemm_blocks, 256, 0, stream>>>(x, W1, t, nullptr, 0, strips);
    k_agg_init <<<(N_NODES * C) / 256, 256, 0, stream>>>(t, dis, b1, out);
    k_agg_edges<<<N_EDGES / 8, 256, 0, stream>>>(ei, t, dis, out);

    // layer 2: t = relu(out) @ W2.T ; out = b2 + self + scatter
    k_gemm_wmma<<<gemm_blocks, 256, 0, stream>>>(out, W2, t, nullptr, 1, strips);
    k_agg_init <<<(N_NODES * C) / 256, 256, 0, stream>>>(t, dis, b2, out);
    k_agg_edges<<<N_EDGES / 8, 256, 0, stream>>>(ei, t, dis, out);

    // final linear (+bias, relu on input) and log_softmax
    k_gemm_wmma<<<gemm_blocks, 256, 0, stream>>>(out, Wl, t, bl, 1, strips);
    k_logsoftmax<<<(N_NODES + 7) / 8, 256, 0, stream>>>(t, out);
}